// VectorQuantizer_56049323213427
// MI455X (gfx1250) — compile-verified
//
#include <hip/hip_runtime.h>
#include <hip/hip_bf16.h>

// ---- problem constants (match reference) ----
#define KCODES 4096
#define DDIM   64
#define NVEC   32768            // 32*1024
#define BETA   0.25f
#define DECAY  0.99f
#define EPSS   1e-5f

typedef float v2f __attribute__((ext_vector_type(2)));
typedef float v8f __attribute__((ext_vector_type(8)));

// ---------------------------------------------------------------------------
// Kernel 1: zero scratch accumulators + precompute ||e||^2 per code.
// grid: K*D threads; first K threads also compute enorm.
// ---------------------------------------------------------------------------
__global__ __launch_bounds__(256) void vq_init(const float* __restrict__ w,
                                               float* __restrict__ enorm,
                                               float* __restrict__ ncs,
                                               float* __restrict__ nes,
                                               float* __restrict__ misc) {
    const int e = blockIdx.x * 256 + threadIdx.x;      // < K*D
    nes[e] = 0.0f;
    if (e < KCODES) {
        ncs[e] = 0.0f;
        const float* wr = w + e * DDIM;
        float s = 0.0f;
#pragma unroll
        for (int d = 0; d < DDIM; d += 4) {
            float4 v = *(const float4*)(wr + d);
            s += v.x * v.x + v.y * v.y + v.z * v.z + v.w * v.w;
        }
        enorm[e] = s;
    }
    if (e == 0) misc[0] = 0.0f;   // loss accumulator
}

// ---------------------------------------------------------------------------
// Kernel 2: fused distance + argmin via V_WMMA_F32_16X16X4_F32.
// Each wave owns 16 rows of x; scans all K codes in 16-wide tiles.
// A (16x4 f32): lane[0..15]=M, VGPR0 holds K=0 (lanes<16) / K=2 (lanes>=16),
// VGPR1 holds K=1 / K=3  ->  per lane one v2f = 8B contiguous load.
// C/D (16x16 f32): VGPR j, lane L -> row M=j+8*(L>=16), col N=L&15.
// ---------------------------------------------------------------------------
__global__ __launch_bounds__(256) void vq_argmin(const float* __restrict__ x,
                                                 const float* __restrict__ w,
                                                 const float* __restrict__ enorm,
                                                 int*   __restrict__ idx,
                                                 float* __restrict__ idx_f) {
    const int lane = threadIdx.x & 31;
    const int wave = threadIdx.x >> 5;
    const int half = lane >> 4;          // 0: K pair {0,1}, 1: K pair {2,3}
    const int l15  = lane & 15;
    const int rowBase = blockIdx.x * 128 + wave * 16;

    // Load this wave's 16x64 slab of x in WMMA-A layout (per-lane registers).
    v2f a[16];
    const float* xr = x + (rowBase + l15) * DDIM + 2 * half;
#pragma unroll
    for (int s = 0; s < 16; ++s)
        a[s] = *(const v2f*)(xr + 4 * s);

    float best[8];
    int   bidx[8];
#pragma unroll
    for (int j = 0; j < 8; ++j) { best[j] = 3.4e38f; bidx[j] = 0; }

    for (int c0 = 0; c0 < KCODES; c0 += 16) {
        const int c = c0 + l15;                       // this lane's column/code
        const float* wr = w + c * DDIM + 2 * half;
        v2f b[16];
#pragma unroll
        for (int s = 0; s < 16; ++s)
            b[s] = *(const v2f*)(wr + 4 * s);

        v8f acc = {};
#pragma unroll
        for (int s = 0; s < 16; ++s)
            acc = __builtin_amdgcn_wmma_f32_16x16x4_f32(
                /*neg_a=*/false, a[s], /*neg_b=*/false, b[s],
                /*c_mod=*/(short)0, acc, /*reuse_a=*/false, /*reuse_b=*/false);

        // dist' = ||e||^2 - 2 x.e  (||x||^2 is row-constant -> argmin invariant)
        const float en = enorm[c];
#pragma unroll
        for (int j = 0; j < 8; ++j) {
            float dlt = en - 2.0f * acc[j];
            if (dlt < best[j]) { best[j] = dlt; bidx[j] = c; }   // '<' keeps lowest c on tie
        }
    }

    // Final min-reduction across the 16 lanes of each half (cols 0..15).
#pragma unroll
    for (int j = 0; j < 8; ++j) {
        float v = best[j];
        int   bi = bidx[j];
#pragma unroll
        for (int m = 1; m <= 8; m <<= 1) {
            float ov  = __shfl_xor(v, m, 32);
            int   obi = __shfl_xor(bi, m, 32);
            if (ov < v || (ov == v && obi < bi)) { v = ov; bi = obi; }
        }
        if (l15 == 0) {
            const int row = rowBase + j + 8 * half;
            idx[row]   = bi;
            idx_f[row] = (float)bi;
        }
    }
}

// ---------------------------------------------------------------------------
// Kernel 3: gather quantized vectors, straight-through output, segment sums
// (fp32 global atomics -> L2 atomic units), commitment-loss partial sums.
// ---------------------------------------------------------------------------
__global__ __launch_bounds__(256) void vq_quantize(const float* __restrict__ x,
                                                   const float* __restrict__ w,
                                                   const int* __restrict__ idx,
                                                   float* __restrict__ q_out,
                                                   float* __restrict__ ncs,
                                                   float* __restrict__ nes,
                                                   float* __restrict__ loss) {
    const int e = blockIdx.x * 256 + threadIdx.x;   // < N*D
    const int n = e >> 6;
    const int d = e & (DDIM - 1);
    const int c = idx[n];
    const float q  = w[c * DDIM + d];
    const float xv = x[e];
    q_out[e] = xv + (q - xv);                        // straight-through value
    const float diff = xv - q;
    float part = diff * diff;
    atomicAdd(&nes[c * DDIM + d], xv);
    if (d == 0) atomicAdd(&ncs[c], 1.0f);
#pragma unroll
    for (int m = 16; m >= 1; m >>= 1) part += __shfl_xor(part, m, 32);
    if ((threadIdx.x & 31) == 0) atomicAdd(loss, part);
}

// ---------------------------------------------------------------------------
// Kernel 4: EMA cluster sizes + block-wide sum n (single 1024-thread block).
// ---------------------------------------------------------------------------
__global__ __launch_bounds__(1024) void vq_finalize_cs(const float* __restrict__ cs_in,
                                                       const float* __restrict__ ncs,
                                                       float* __restrict__ cs_out,
                                                       float* __restrict__ nsum) {
    __shared__ float red[32];
    const int tid = threadIdx.x;
    float local = 0.0f;
#pragma unroll
    for (int i = 0; i < KCODES; i += 1024) {
        const float v = DECAY * cs_in[i + tid] + (1.0f - DECAY) * ncs[i + tid];
        cs_out[i + tid] = v;
        local += v;
    }
#pragma unroll
    for (int m = 16; m >= 1; m >>= 1) local += __shfl_xor(local, m, 32);
    if ((tid & 31) == 0) red[tid >> 5] = local;
    __syncthreads();
    if (tid < 32) {
        float v = red[tid];
#pragma unroll
        for (int m = 16; m >= 1; m >>= 1) v += __shfl_xor(v, m, 32);
        if (tid == 0) nsum[0] = v;
    }
}

// ---------------------------------------------------------------------------
// Kernel 5: EMA embedding sums, smoothed new_weight, final loss scalar.
// ---------------------------------------------------------------------------
__global__ __launch_bounds__(256) void vq_finalize(const float* __restrict__ es_in,
                                                   const float* __restrict__ nes,
                                                   const float* __restrict__ cs_out,
                                                   const float* __restrict__ nsum,
                                                   const float* __restrict__ lossacc,
                                                   float* __restrict__ w_out,
                                                   float* __restrict__ es_out,
                                                   float* __restrict__ loss_out) {
    const int e = blockIdx.x * 256 + threadIdx.x;    // < K*D
    const int c = e >> 6;
    const float es = DECAY * es_in[e] + (1.0f - DECAY) * nes[e];
    es_out[e] = es;
    const float n  = nsum[0];
    const float cs = cs_out[c];
    const float csm = (cs + EPSS) / (n + (float)KCODES * EPSS) * n;
    w_out[e] = es / csm;
    if (e == 0) loss_out[0] = BETA * lossacc[0] / (float)(NVEC * DDIM);
}

// ---------------------------------------------------------------------------
// Launch. Inputs: [0]=inputs(32,1024,64) [1]=weight(K,D) [2]=ema_cs(K)
// [3]=ema_es(K,D). Output (flat f32, return order):
// quantized_st | loss | indices | new_weight | ema_cs | ema_es
// ---------------------------------------------------------------------------
extern "C" void kernel_launch(void* const* d_in, const int* in_sizes, int n_in,
                              void* d_out, int out_size, void* d_ws, size_t ws_size,
                              hipStream_t stream) {
    const float* x     = (const float*)d_in[0];
    const float* w     = (const float*)d_in[1];
    const float* cs_in = (const float*)d_in[2];
    const float* es_in = (const float*)d_in[3];

    float* out      = (float*)d_out;
    float* out_q    = out;                                   // 2097152
    float* out_loss = out + 2097152;                         // 1
    float* out_idx  = out + 2097153;                         // 32768
    float* out_w    = out + 2129921;                         // 262144
    float* out_cs   = out + 2392065;                         // 4096
    float* out_es   = out + 2396161;                         // 262144

    float* ws       = (float*)d_ws;
    float* ws_enorm = ws;                                    // K
    int*   ws_idx   = (int*)(ws + 4096);                     // N ints
    float* ws_ncs   = ws + 36864;                            // K
    float* ws_nes   = ws + 40960;                            // K*D
    float* ws_misc  = ws + 303104;                           // [0]=loss, [1]=n

    vq_init       <<<(KCODES * DDIM) / 256, 256, 0, stream>>>(w, ws_enorm, ws_ncs, ws_nes, ws_misc);
    vq_argmin     <<<NVEC / 128,           256, 0, stream>>>(x, w, ws_enorm, ws_idx, out_idx);
    vq_quantize   <<<(NVEC * DDIM) / 256,  256, 0, stream>>>(x, w, ws_idx, out_q, ws_ncs, ws_nes, ws_misc);
    vq_finalize_cs<<<1,                   1024, 0, stream>>>(cs_in, ws_ncs, out_cs, ws_misc + 1);
    vq_finalize   <<<(KCODES * DDIM) / 256, 256, 0, stream>>>(es_in, ws_nes, out_cs, ws_misc + 1,
                                                              ws_misc, out_w, out_es, out_loss);
}